// Encoder_34565896798379
// MI455X (gfx1250) — compile-verified
//
#include <hip/hip_runtime.h>

typedef __attribute__((ext_vector_type(2))) float v2f;
typedef __attribute__((ext_vector_type(8))) float v8f;

#define NQ 16
#define NCOL (1u << NQ)          // 65536 columns (2^16 amplitudes)
#define TILES_PER_WAVE 8
#define WAVES_PER_BLOCK 8
#define THREADS 256

// Phase-GEMM via V_WMMA_F32_16X16X4_F32:
//   Theta(256 x 65536) = X(256 x 16) * S(16 x 65536),  S[q][k] = bit_q(k) ? +pi : -pi
// then out[b,k] = (cos(theta), sin(theta)) / 256.
__global__ __launch_bounds__(THREADS)
void qenc_wmma_kernel(const float* __restrict__ x, float* __restrict__ out) {
    const int lane = threadIdx.x & 31;
    const int wave = blockIdx.x * WAVES_PER_BLOCK + (threadIdx.x >> 5);

    // 4096 n-tiles per m-tile, 8 per wave -> 512 waves per m-tile, 16 m-tiles.
    const int wavesPerM = (NCOL / 16) / TILES_PER_WAVE;   // 512
    const int mTile = wave / wavesPerM;                   // 0..15
    const int chunk = wave % wavesPerM;                   // 0..511
    const int m0    = mTile * 16;

    const int lm   = lane & 15;   // col (for B/C/D) or row (for A)
    const int hi   = lane >> 4;   // half-wave select
    const int koff = hi * 2;      // A/B K offset within a K=4 group

    // A-matrix (16x4 f32, ISA layout): lane L<16 holds {K0,K1} of row L,
    // lane L>=16 holds {K2,K3} of row L-16. Four K-groups cover K=16 qubits.
    const float* arow = x + (m0 + lm) * NQ + koff;
    v2f a[4];
#pragma unroll
    for (int i = 0; i < 4; ++i)
        a[i] = *(const v2f*)(arow + 4 * i);

    const float PI_F  = 3.14159265358979323846f;
    const float SCALE = 0.00390625f;              // (1/sqrt(2))^16 = 2^-8

    for (int tt = 0; tt < TILES_PER_WAVE; ++tt) {
        const unsigned t    = (unsigned)(chunk * TILES_PER_WAVE + tt); // n-tile 0..4095
        const unsigned kcol = (t << 4) | (unsigned)lm;                 // global column

        v8f acc = {};   // theta tile, 16x16 f32
#pragma unroll
        for (int i = 0; i < 4; ++i) {
            // B-matrix (4x16 f32): mirror of A layout; this lane supplies
            // sign rows (qubits) q0 and q0+1 for column lm.
            const int q0 = 4 * i + koff;
            v2f b;
            b.x = ((kcol >> (15 - q0)) & 1u) ? PI_F : -PI_F;
            b.y = ((kcol >> (14 - q0)) & 1u) ? PI_F : -PI_F;
            acc = __builtin_amdgcn_wmma_f32_16x16x4_f32(
                false, a[i], false, b, (short)0, acc, false, false);
        }

        // C/D layout: VGPR r -> row m0 + r + 8*hi, lane lm -> col.
        // Lanes 0..15 of each VGPR store 16 consecutive complex values (128B).
#pragma unroll
        for (int r = 0; r < 8; ++r) {
            const float theta = acc[r];
            const unsigned row = (unsigned)(m0 + r + 8 * hi);
            const unsigned idx = (row << NQ) | kcol;   // flat [256, 65536] index
            v2f o;
            o.x = __cosf(theta) * SCALE;               // Re
            o.y = __sinf(theta) * SCALE;               // Im
            *(v2f*)(out + 2ull * idx) = o;
        }
    }
}

extern "C" void kernel_launch(void* const* d_in, const int* in_sizes, int n_in,
                              void* d_out, int out_size, void* d_ws, size_t ws_size,
                              hipStream_t stream) {
    const float* x = (const float*)d_in[0];   // [256, 16] float32
    float* out = (float*)d_out;               // [256, 65536] complex64 -> interleaved f32

    // 16 m-tiles * 512 chunks = 8192 waves = 1024 blocks of 8 waves.
    dim3 grid(1024), block(THREADS);
    hipLaunchKernelGGL(qenc_wmma_kernel, grid, block, 0, stream, x, out);
}